// L1_66090956751452
// MI455X (gfx1250) — compile-verified
//
#include <hip/hip_runtime.h>

// CDNA5 / gfx1250, wave32. One workgroup (256 thr = 8 waves) per batch element.
// Fused 3-stage Tucker transform entirely in LDS (2 x 128KB ping-pong buffers
// out of the WGP's 320KB), f32 WMMA 16x16x4 for all matrix math.

typedef float v2f __attribute__((ext_vector_type(2)));
typedef float v8f __attribute__((ext_vector_type(8)));

#define FDIM  32
#define RROWS 1024   // 32*32 rows per stage-GEMM
#define ELEMS 32768  // 32^3 floats per batch element

// One stage:  OUT[r*32 + a] = sum_c IN[c*1024 + r] * W[c*32 + a]
// M-tiles: 64 (this wave does 8), N-tiles: 2, K-chunks: 8 (K=4 each).
#define STAGE(INPTR, WPTR, OUTPTR)                                            \
  {                                                                           \
    /* Preload all B fragments for this stage (shared across M-tiles). */     \
    v2f bfrag[2][8];                                                          \
    _Pragma("unroll")                                                         \
    for (int nt = 0; nt < 2; ++nt) {                                          \
      _Pragma("unroll")                                                       \
      for (int kc = 0; kc < 8; ++kc) {                                        \
        const int k = kc * 4 + laneH * 2;                                     \
        const int n = nt * 16 + laneM;                                        \
        bfrag[nt][kc].x = (WPTR)[k * FDIM + n];                               \
        bfrag[nt][kc].y = (WPTR)[(k + 1) * FDIM + n];                         \
      }                                                                       \
    }                                                                         \
    _Pragma("unroll")                                                         \
    for (int i = 0; i < 8; ++i) {                                             \
      const int r0 = (wave * 8 + i) * 16;                                     \
      v2f afrag[8];                                                           \
      _Pragma("unroll")                                                       \
      for (int kc = 0; kc < 8; ++kc) {                                        \
        const int c = kc * 4 + laneH * 2;                                     \
        const int r = r0 + laneM;                                             \
        afrag[kc].x = (INPTR)[c * RROWS + r];                                 \
        afrag[kc].y = (INPTR)[(c + 1) * RROWS + r];                           \
      }                                                                       \
      v8f acc0 = {0.f, 0.f, 0.f, 0.f, 0.f, 0.f, 0.f, 0.f};                    \
      v8f acc1 = {0.f, 0.f, 0.f, 0.f, 0.f, 0.f, 0.f, 0.f};                    \
      _Pragma("unroll")                                                       \
      for (int kc = 0; kc < 8; ++kc) {                                        \
        acc0 = __builtin_amdgcn_wmma_f32_16x16x4_f32(                         \
            false, afrag[kc], false, bfrag[0][kc], (short)0, acc0,            \
            false, false);                                                    \
        acc1 = __builtin_amdgcn_wmma_f32_16x16x4_f32(                         \
            false, afrag[kc], false, bfrag[1][kc], (short)0, acc1,            \
            false, false);                                                    \
      }                                                                       \
      /* D layout: vgpr v, lane L -> (M = v + 8*(L>>4), N = L&15) */          \
      _Pragma("unroll")                                                       \
      for (int v = 0; v < 8; ++v) {                                           \
        const int r = r0 + v + laneH * 8;                                     \
        (OUTPTR)[r * FDIM + laneM]      = acc0[v];                            \
        (OUTPTR)[r * FDIM + 16 + laneM] = acc1[v];                            \
      }                                                                       \
    }                                                                         \
  }

__global__ __launch_bounds__(256) void kron3_wmma_kernel(
    const float* __restrict__ x,
    const float* __restrict__ w0,
    const float* __restrict__ w1,
    const float* __restrict__ w2,
    float* __restrict__ out)
{
  extern __shared__ float smem[];
  float* buf0 = smem;          // 32768 floats (y1)
  float* buf1 = smem + ELEMS;  // 32768 floats (y2)

  const int tid   = threadIdx.x;
  const int wave  = tid >> 5;   // 0..7
  const int lane  = tid & 31;
  const int laneM = lane & 15;  // M / N position within fragment
  const int laneH = lane >> 4;  // half-wave selector (K/M offset)

  const long long base = (long long)blockIdx.x * (long long)ELEMS;
  const float* xb = x + base;
  float* ob = out + base;

  // Stage 1: global -> LDS   (contract f1)
  STAGE(xb, w0, buf0);
  __syncthreads();
  // Stage 2: LDS -> LDS      (contract f2)
  STAGE(buf0, w1, buf1);
  __syncthreads();
  // Stage 3: LDS -> global   (contract f3)
  STAGE(buf1, w2, ob);
}

extern "C" void kernel_launch(void* const* d_in, const int* in_sizes, int n_in,
                              void* d_out, int out_size, void* d_ws, size_t ws_size,
                              hipStream_t stream) {
  const float* x  = (const float*)d_in[0];
  const float* w0 = (const float*)d_in[1];
  const float* w1 = (const float*)d_in[2];
  const float* w2 = (const float*)d_in[3];
  float* out = (float*)d_out;

  const int batch = in_sizes[0] / ELEMS;           // 1024
  const size_t shmem = 2 * ELEMS * sizeof(float);  // 256 KB of the 320 KB WGP LDS

  kron3_wmma_kernel<<<batch, 256, shmem, stream>>>(x, w0, w1, w2, out);
}